// DGCNN2D_54511724921220
// MI455X (gfx1250) — compile-verified
//
#include <hip/hip_runtime.h>
#include <hip/hip_bf16.h>

#define KNN 20
#define NPTS 2048
#define NB 8
#define EPS 1e-5f

typedef __attribute__((ext_vector_type(2))) float v2f;
typedef __attribute__((ext_vector_type(8))) float v8f;

#define DEVINL __device__ __forceinline__

static DEVINL float lrelu_f(float x) { return x >= 0.f ? x : 0.2f * x; }

// order-preserving float<->uint encoding for atomicMax on floats
static DEVINL unsigned fenc(float f) {
    unsigned u = __float_as_uint(f);
    return (u & 0x80000000u) ? ~u : (u | 0x80000000u);
}
static DEVINL float fdec(unsigned u) {
    return (u & 0x80000000u) ? __uint_as_float(u & 0x7fffffffu) : __uint_as_float(~u);
}

// ---------------------------------------------------------------------------
// KNN: one wave (32 lanes) per row n. Per-lane sorted top-20 over 64
// candidates, then 20 rounds of wave argmax via shuffles.
// xs layout: [b][c][n], channel stride NPTS, batch stride = batchStride.
// ---------------------------------------------------------------------------
template<int C>
__global__ void knn_kernel(const float* __restrict__ xs, int batchStride,
                           int* __restrict__ idxOut)
{
    const int lane = threadIdx.x & 31;
    const int wave = threadIdx.x >> 5;
    const int row  = blockIdx.x * 8 + wave;        // over B*N
    const int b = row >> 11;
    const int n = row & (NPTS - 1);
    const float* xb = xs + (long)b * batchStride;

    float p[C];
    float sqn = 0.f;
#pragma unroll
    for (int c = 0; c < C; ++c) { p[c] = xb[c * NPTS + n]; sqn += p[c] * p[c]; }

    float dk[KNN]; int ik[KNN];
#pragma unroll
    for (int t = 0; t < KNN; ++t) { dk[t] = -3.4e38f; ik[t] = 0; }

    for (int mb = 0; mb < NPTS; mb += 32) {
        int m = mb + lane;
        float inner = 0.f, sqm = 0.f;
#pragma unroll
        for (int c = 0; c < C; ++c) {
            float q = xb[c * NPTS + m];
            inner += p[c] * q; sqm += q * q;
        }
        float nd = 2.f * inner - sqn - sqm;
        if (nd > dk[KNN - 1]) {
            dk[KNN - 1] = nd; ik[KNN - 1] = m;
#pragma unroll
            for (int t = KNN - 1; t > 0; --t) {
                if (dk[t] > dk[t - 1]) {
                    float td = dk[t]; dk[t] = dk[t - 1]; dk[t - 1] = td;
                    int ti = ik[t]; ik[t] = ik[t - 1]; ik[t - 1] = ti;
                }
            }
        }
    }

    // merge 32 sorted lists: 20 rounds of wave argmax
    float cv = dk[0]; int ci = ik[0];
    int* out = idxOut + (long)row * KNN;
    for (int r = 0; r < KNN; ++r) {
        float bv = cv; int bl = lane;
        for (int off = 16; off; off >>= 1) {
            float ov = __shfl_xor(bv, off);
            int   ol = __shfl_xor(bl, off);
            if (ov > bv || (ov == bv && ol < bl)) { bv = ov; bl = ol; }
        }
        int widx = __shfl(ci, bl);
        if (lane == 0) out[r] = widx;
        if (lane == bl) {
#pragma unroll
            for (int t = 0; t < KNN - 1; ++t) { dk[t] = dk[t + 1]; ik[t] = ik[t + 1]; }
            dk[KNN - 1] = -3.4e38f;
            cv = dk[0]; ci = ik[0];
        }
    }
}

// ---------------------------------------------------------------------------
// Edge-conv GEMM, fused with BN-stat accumulation and max-over-k.
// D(O x 20) per point = W(O x 2C) * E(2C x 20); E built on the fly:
//   E[c][k]   = x[c][nbr_k] - x[c][n]   (c <  C)
//   E[C+c][k] = x[c][n]                 (c >= C)
// One wave per point n; two 16-col WMMA tiles cover k = 0..19.
// Block = 4 waves (same o-tile), LDS-combined stats.
// ---------------------------------------------------------------------------
template<int C>
DEVINL float evalE(const float* __restrict__ xb, int c, int n, int nbr) {
    return (c < C) ? (xb[c * NPTS + nbr] - xb[c * NPTS + n])
                   : xb[(c - C) * NPTS + n];
}

template<int C, int O>
__global__ void edge_gemm_kernel(const float* __restrict__ xs, int batchStride,
                                 const int* __restrict__ idx,
                                 const float* __restrict__ W,   // O x 2C row-major
                                 float* __restrict__ xmax,      // B x O x N (raw max over k)
                                 float* __restrict__ stats)     // [0,O) sum, [O,2O) sumsq
{
    constexpr int CK = 2 * C;
    __shared__ float s_sum[16], s_ssq[16];
    const int lane = threadIdx.x & 31;
    const int wave = threadIdx.x >> 5;
    if (threadIdx.x < 16) { s_sum[threadIdx.x] = 0.f; s_ssq[threadIdx.x] = 0.f; }
    __syncthreads();

    const int n  = blockIdx.x * 4 + wave;
    const int ot = blockIdx.y;
    const int b  = blockIdx.z;
    const float* xb = xs + (long)b * batchStride;
    const int* nbrs = idx + ((long)b * NPTS + n) * KNN;

    const int col = lane & 15;     // N index within tile
    const int grp = lane >> 4;     // K half
    const int o_row = ot * 16 + col;

    int nb1 = nbrs[col];                              // k = col      (all valid)
    int nb2 = (col < 4) ? nbrs[16 + col] : n;         // k = 16+col   (valid col<4)

    v8f acc1 = {}; v8f acc2 = {};
#pragma unroll
    for (int kb = 0; kb < CK; kb += 4) {
        int c0 = kb + 2 * grp;
        int c1 = c0 + 1;
        v2f a, bf1, bf2;
        a[0] = W[o_row * CK + c0];
        a[1] = W[o_row * CK + c1];
        bf1[0] = evalE<C>(xb, c0, n, nb1);
        bf1[1] = evalE<C>(xb, c1, n, nb1);
        bf2[0] = evalE<C>(xb, c0, n, nb2);
        bf2[1] = evalE<C>(xb, c1, n, nb2);
        acc1 = __builtin_amdgcn_wmma_f32_16x16x4_f32(false, a, false, bf1,
                                                     (short)0, acc1, false, false);
        acc2 = __builtin_amdgcn_wmma_f32_16x16x4_f32(false, a, false, bf2,
                                                     (short)0, acc2, false, false);
    }

    const bool valid2 = (col < 4);
#pragma unroll
    for (int r = 0; r < 8; ++r) {
        float v1 = acc1[r];
        float v2 = valid2 ? acc2[r] : -3.4e38f;
        float cand = fmaxf(v1, v2);
        for (int off = 8; off; off >>= 1)              // within 16-lane half
            cand = fmaxf(cand, __shfl_xor(cand, off));
        int o = ot * 16 + r + 8 * grp;
        if (col == 0) xmax[((long)b * O + o) * NPTS + n] = cand;
        float sv = v1 + (valid2 ? acc2[r] : 0.f);
        float sq = v1 * v1 + (valid2 ? acc2[r] * acc2[r] : 0.f);
        atomicAdd(&s_sum[r + 8 * grp], sv);
        atomicAdd(&s_ssq[r + 8 * grp], sq);
    }
    __syncthreads();
    if (threadIdx.x < 16) {
        atomicAdd(&stats[ot * 16 + threadIdx.x], s_sum[threadIdx.x]);
        atomicAdd(&stats[O + ot * 16 + threadIdx.x], s_ssq[threadIdx.x]);
    }
}

// finish stage: hcat[b][chOff+o][n] = lrelu(BN(xmax))
__global__ void bn_finish_kernel(const float* __restrict__ xmax,
                                 const float* __restrict__ stats,
                                 const float* __restrict__ g,
                                 const float* __restrict__ bb,
                                 float* __restrict__ hcat,
                                 int O, int chOff, float invCnt)
{
    int i = blockIdx.x * blockDim.x + threadIdx.x;
    if (i >= NB * O * NPTS) return;
    int n = i & (NPTS - 1);
    int o = (i >> 11) % O;
    int b = i / (O * NPTS);
    float mean = stats[o] * invCnt;
    float var  = stats[O + o] * invCnt - mean * mean;
    float rstd = rsqrtf(var + EPS);
    float v = (xmax[i] - mean) * rstd * g[o] + bb[o];
    hcat[((long)b * 112 + chOff + o) * NPTS + n] = lrelu_f(v);
}

// ---------------------------------------------------------------------------
// w4 stage: y = W4(128x112) * hcat, fused stats + half-pool (max over each
// 1024-point half) via order-preserving uint atomicMax.
// ---------------------------------------------------------------------------
__global__ void gemm_w4_kernel(const float* __restrict__ h,   // B x 112 x N
                               const float* __restrict__ W,   // 128 x 112
                               unsigned* __restrict__ pmax,   // B x 128 x 2 encoded
                               float* __restrict__ stats)     // 128 sum + 128 ssq
{
    __shared__ float s_sum[16], s_ssq[16];
    const int lane = threadIdx.x & 31;
    const int wave = threadIdx.x >> 5;
    if (threadIdx.x < 16) { s_sum[threadIdx.x] = 0.f; s_ssq[threadIdx.x] = 0.f; }
    __syncthreads();

    const int ot = blockIdx.y;
    const int b  = blockIdx.z;
    const int col = lane & 15;
    const int grp = lane >> 4;
    const int n0 = (blockIdx.x * 4 + wave) * 16;
    const int n  = n0 + col;
    const int half = n0 >> 10;     // uniform within tile
    const float* hb = h + (long)b * 112 * NPTS;

    v8f acc = {};
#pragma unroll 4
    for (int kb = 0; kb < 112; kb += 4) {
        int c0 = kb + 2 * grp;
        v2f a, bf;
        a[0] = W[(ot * 16 + col) * 112 + c0];
        a[1] = W[(ot * 16 + col) * 112 + c0 + 1];
        bf[0] = hb[c0 * NPTS + n];
        bf[1] = hb[(c0 + 1) * NPTS + n];
        acc = __builtin_amdgcn_wmma_f32_16x16x4_f32(false, a, false, bf,
                                                    (short)0, acc, false, false);
    }

#pragma unroll
    for (int r = 0; r < 8; ++r) {
        float v = acc[r];
        atomicAdd(&s_sum[r + 8 * grp], v);
        atomicAdd(&s_ssq[r + 8 * grp], v * v);
        float m = v;
        for (int off = 8; off; off >>= 1) m = fmaxf(m, __shfl_xor(m, off));
        if (col == 0) {
            int o = ot * 16 + r + 8 * grp;
            atomicMax(&pmax[((long)b * 128 + o) * 2 + half], fenc(m));
        }
    }
    __syncthreads();
    if (threadIdx.x < 16) {
        atomicAdd(&stats[ot * 16 + threadIdx.x], s_sum[threadIdx.x]);
        atomicAdd(&stats[128 + ot * 16 + threadIdx.x], s_ssq[threadIdx.x]);
    }
}

__global__ void bn4_finish_kernel(const unsigned* __restrict__ pmax,
                                  const float* __restrict__ stats,
                                  const float* __restrict__ g4,
                                  const float* __restrict__ b4,
                                  float* __restrict__ pooled)   // B x 128 x 2
{
    int i = blockIdx.x * blockDim.x + threadIdx.x;
    if (i >= NB * 128 * 2) return;
    int o = (i >> 1) & 127;
    const float invCnt = 1.f / (float)(NB * NPTS);
    float mean = stats[o] * invCnt;
    float var  = stats[128 + o] * invCnt - mean * mean;
    float rstd = rsqrtf(var + EPS);
    float v = (fdec(pmax[i]) - mean) * rstd * g4[o] + b4[o];
    pooled[i] = lrelu_f(v);
}

// ---------------------------------------------------------------------------
// tiny dense layers (w5, w6): cols = 16 (B*2)
// ---------------------------------------------------------------------------
__global__ void dense_kernel(const float* __restrict__ in,  // B x Cin x 2
                             const float* __restrict__ W,   // O x Cin
                             const float* __restrict__ cb,  // O
                             float* __restrict__ z,         // O x 16
                             int Cin, int O)
{
    int t = blockIdx.x * blockDim.x + threadIdx.x;
    if (t >= O * 16) return;
    int o = t >> 4; int col = t & 15; int b = col >> 1; int j = col & 1;
    float s = cb[o];
    for (int c = 0; c < Cin; ++c)
        s += W[o * Cin + c] * in[((long)b * Cin + c) * 2 + j];
    z[o * 16 + col] = s;
}

__global__ void bn_relu16_kernel(const float* __restrict__ z,
                                 const float* __restrict__ g,
                                 const float* __restrict__ bb,
                                 float* __restrict__ out,  // B x O x 2
                                 int O)
{
    int o = blockIdx.x * blockDim.x + threadIdx.x;
    if (o >= O) return;
    float s = 0.f, ss = 0.f;
    for (int c = 0; c < 16; ++c) { float v = z[o * 16 + c]; s += v; ss += v * v; }
    float mean = s / 16.f;
    float var  = ss / 16.f - mean * mean;
    float rstd = rsqrtf(var + EPS);
    for (int c = 0; c < 16; ++c) {
        float v = (z[o * 16 + c] - mean) * rstd * g[o] + bb[o];
        int b = c >> 1, j = c & 1;
        out[((long)b * O + o) * 2 + j] = fmaxf(v, 0.f);
    }
}

// h6 (B,256,2) -> g0 (B,2,16,16)
__global__ void remap_kernel(const float* __restrict__ h6, float* __restrict__ g0)
{
    int i = blockIdx.x * blockDim.x + threadIdx.x;
    if (i >= NB * 512) return;
    int b = i >> 9; int lin = i & 511;
    int o = lin >> 1; int j = lin & 1;
    g0[i] = h6[((long)b * 256 + o) * 2 + j];
}

// ---------------------------------------------------------------------------
// transposed conv: stride 2, kernel 4, pad 2: (B,Ci,H,H) -> (B,Co,2H,2H)
// w layout: (Ci, Co, 4, 4);  wf[o][i][ky][kx] = w[i][o][3-ky][3-kx]
// ---------------------------------------------------------------------------
__global__ void deconv_kernel(const float* __restrict__ in,
                              const float* __restrict__ w,
                              const float* __restrict__ bias,
                              float* __restrict__ out,
                              int Ci, int Co, int H)
{
    int Ho = 2 * H;
    int total = NB * Co * Ho * Ho;
    int i = blockIdx.x * blockDim.x + threadIdx.x;
    if (i >= total) return;
    int ox = i % Ho;
    int oy = (i / Ho) % Ho;
    int o  = (i / (Ho * Ho)) % Co;
    int b  = i / (Co * Ho * Ho);
    float s = bias[o];
    for (int ci = 0; ci < Ci; ++ci) {
        const float* inp = in + ((long)b * Ci + ci) * H * H;
        const float* wp = w + ((long)ci * Co + o) * 16;
#pragma unroll
        for (int ky = 0; ky < 4; ++ky) {
            int iy2 = oy + ky - 2;
            if (iy2 < 0 || (iy2 & 1) || iy2 > 2 * H - 2) continue;
            int iy = iy2 >> 1;
#pragma unroll
            for (int kx = 0; kx < 4; ++kx) {
                int ix2 = ox + kx - 2;
                if (ix2 < 0 || (ix2 & 1) || ix2 > 2 * H - 2) continue;
                int ix = ix2 >> 1;
                s += wp[(3 - ky) * 4 + (3 - kx)] * inp[iy * H + ix];
            }
        }
    }
    out[i] = s;
}

// instance-norm + lrelu in place; one block per (b,c)
__global__ void inorm_lrelu_kernel(float* __restrict__ buf, int HW)
{
    __shared__ float ssum[256], sssq[256];
    float* p = buf + (long)blockIdx.x * HW;
    int tid = threadIdx.x;
    float s = 0.f, ss = 0.f;
    for (int i = tid; i < HW; i += 256) { float v = p[i]; s += v; ss += v * v; }
    ssum[tid] = s; sssq[tid] = ss;
    __syncthreads();
    for (int off = 128; off; off >>= 1) {
        if (tid < off) { ssum[tid] += ssum[tid + off]; sssq[tid] += sssq[tid + off]; }
        __syncthreads();
    }
    float mean = ssum[0] / (float)HW;
    float var  = sssq[0] / (float)HW - mean * mean;
    float rstd = rsqrtf(var + EPS);
    for (int i = tid; i < HW; i += 256)
        p[i] = lrelu_f((p[i] - mean) * rstd);
}

// ---------------------------------------------------------------------------
extern "C" void kernel_launch(void* const* d_in, const int* in_sizes, int n_in,
                              void* d_out, int out_size, void* d_ws, size_t ws_size,
                              hipStream_t stream)
{
    (void)in_sizes; (void)n_in; (void)out_size; (void)ws_size;
    const float* x   = (const float*)d_in[0];
    const float* w1  = (const float*)d_in[1];
    const float* g1  = (const float*)d_in[2];
    const float* b1  = (const float*)d_in[3];
    const float* w2  = (const float*)d_in[4];
    const float* g2  = (const float*)d_in[5];
    const float* b2  = (const float*)d_in[6];
    const float* w3  = (const float*)d_in[7];
    const float* g3  = (const float*)d_in[8];
    const float* b3  = (const float*)d_in[9];
    const float* w4  = (const float*)d_in[10];
    const float* g4  = (const float*)d_in[11];
    const float* b4  = (const float*)d_in[12];
    const float* w5  = (const float*)d_in[13];
    const float* c5  = (const float*)d_in[14];
    const float* g5  = (const float*)d_in[15];
    const float* b5  = (const float*)d_in[16];
    const float* w6  = (const float*)d_in[17];
    const float* c6  = (const float*)d_in[18];
    const float* g6  = (const float*)d_in[19];
    const float* b6  = (const float*)d_in[20];
    const float* dw0 = (const float*)d_in[21];
    const float* db0 = (const float*)d_in[22];
    const float* dw1 = (const float*)d_in[23];
    const float* db1 = (const float*)d_in[24];
    const float* dw2 = (const float*)d_in[25];
    const float* db2 = (const float*)d_in[26];
    const float* dw3 = (const float*)d_in[27];
    const float* db3 = (const float*)d_in[28];
    float* outp = (float*)d_out;

    // workspace carve-up (all 4-byte elements)
    size_t off = 0;
    char* wsb = (char*)d_ws;
    int*      idx    = (int*)(wsb + off);      off += (size_t)NB * NPTS * KNN * 4;
    float*    hcat   = (float*)(wsb + off);    off += (size_t)NB * 112 * NPTS * 4;
    float*    xmax   = (float*)(wsb + off);    off += (size_t)NB * 64 * NPTS * 4;
    float*    stats  = (float*)(wsb + off);    off += 256 * 4;
    unsigned* pmax   = (unsigned*)(wsb + off); off += (size_t)NB * 128 * 2 * 4;
    float*    pooled = (float*)(wsb + off);    off += (size_t)NB * 128 * 2 * 4;
    float*    z5     = (float*)(wsb + off);    off += 256 * 16 * 4;
    float*    h5     = (float*)(wsb + off);    off += (size_t)NB * 256 * 2 * 4;
    float*    z6     = (float*)(wsb + off);    off += 256 * 16 * 4;
    float*    h6     = (float*)(wsb + off);    off += (size_t)NB * 256 * 2 * 4;
    float*    g0     = (float*)(wsb + off);    off += (size_t)NB * 2 * 16 * 16 * 4;
    float*    t1     = (float*)(wsb + off);    off += (size_t)NB * 2 * 32 * 32 * 4;
    float*    t2     = (float*)(wsb + off);    off += (size_t)NB * 2 * 64 * 64 * 4;
    float*    t3     = (float*)(wsb + off);    off += (size_t)NB * 2 * 128 * 128 * 4;

    const int rowBlocks = NB * NPTS / 8;
    const dim3 egrid(NPTS / 4, 1, NB);

    // ---- stage 1: C=2, O=16 ----
    hipMemsetAsync(stats, 0, 256 * 4, stream);
    knn_kernel<2><<<rowBlocks, 256, 0, stream>>>(x, 2 * NPTS, idx);
    edge_gemm_kernel<2, 16><<<dim3(NPTS / 4, 1, NB), 128, 0, stream>>>(
        x, 2 * NPTS, idx, w1, xmax, stats);
    bn_finish_kernel<<<(NB * 16 * NPTS + 255) / 256, 256, 0, stream>>>(
        xmax, stats, g1, b1, hcat, 16, 0, 1.f / (float)(NB * NPTS * KNN));

    // ---- stage 2: C=16 (hcat ch0..15), O=32 ----
    hipMemsetAsync(stats, 0, 256 * 4, stream);
    knn_kernel<16><<<rowBlocks, 256, 0, stream>>>(hcat, 112 * NPTS, idx);
    edge_gemm_kernel<16, 32><<<dim3(NPTS / 4, 2, NB), 128, 0, stream>>>(
        hcat, 112 * NPTS, idx, w2, xmax, stats);
    bn_finish_kernel<<<(NB * 32 * NPTS + 255) / 256, 256, 0, stream>>>(
        xmax, stats, g2, b2, hcat, 32, 16, 1.f / (float)(NB * NPTS * KNN));

    // ---- stage 3: C=32 (hcat ch16..47), O=64 ----
    hipMemsetAsync(stats, 0, 256 * 4, stream);
    knn_kernel<32><<<rowBlocks, 256, 0, stream>>>(hcat + 16 * NPTS, 112 * NPTS, idx);
    edge_gemm_kernel<32, 64><<<dim3(NPTS / 4, 4, NB), 128, 0, stream>>>(
        hcat + 16 * NPTS, 112 * NPTS, idx, w3, xmax, stats);
    bn_finish_kernel<<<(NB * 64 * NPTS + 255) / 256, 256, 0, stream>>>(
        xmax, stats, g3, b3, hcat, 64, 48, 1.f / (float)(NB * NPTS * KNN));

    // ---- w4 + BN + lrelu + half-pool ----
    hipMemsetAsync(stats, 0, 256 * 4, stream);
    hipMemsetAsync(pmax, 0, (size_t)NB * 128 * 2 * 4, stream);
    gemm_w4_kernel<<<dim3(NPTS / 64, 8, NB), 128, 0, stream>>>(hcat, w4, pmax, stats);
    bn4_finish_kernel<<<(NB * 128 * 2 + 255) / 256, 256, 0, stream>>>(
        pmax, stats, g4, b4, pooled);

    // ---- w5, w6 (tiny) ----
    dense_kernel<<<(256 * 16 + 255) / 256, 256, 0, stream>>>(pooled, w5, c5, z5, 128, 256);
    bn_relu16_kernel<<<1, 256, 0, stream>>>(z5, g5, b5, h5, 256);
    dense_kernel<<<(256 * 16 + 255) / 256, 256, 0, stream>>>(h5, w6, c6, z6, 256, 256);
    bn_relu16_kernel<<<1, 256, 0, stream>>>(z6, g6, b6, h6, 256);

    // ---- reshape + deconv chain ----
    remap_kernel<<<(NB * 512 + 255) / 256, 256, 0, stream>>>(h6, g0);

    deconv_kernel<<<(NB * 2 * 32 * 32 + 255) / 256, 256, 0, stream>>>(g0, dw0, db0, t1, 2, 2, 16);
    inorm_lrelu_kernel<<<NB * 2, 256, 0, stream>>>(t1, 32 * 32);

    deconv_kernel<<<(NB * 2 * 64 * 64 + 255) / 256, 256, 0, stream>>>(t1, dw1, db1, t2, 2, 2, 32);
    inorm_lrelu_kernel<<<NB * 2, 256, 0, stream>>>(t2, 64 * 64);

    deconv_kernel<<<(NB * 2 * 128 * 128 + 255) / 256, 256, 0, stream>>>(t2, dw2, db2, t3, 2, 2, 64);
    inorm_lrelu_kernel<<<NB * 2, 256, 0, stream>>>(t3, 128 * 128);

    deconv_kernel<<<(NB * 1 * 256 * 256 + 255) / 256, 256, 0, stream>>>(t3, dw3, db3, outp, 2, 1, 128);
}